// Brain_connectomic_graph_52226802319568
// MI455X (gfx1250) — compile-verified
//
#include <hip/hip_runtime.h>
#include <math.h>

#define NN 227328
#define EE 1000000
#define HH 128
#define NB_ 111
#define PP_ 40
#define MAXLEN_ 10
#define NSTART_ 5

typedef __attribute__((ext_vector_type(2))) float v2f;
typedef __attribute__((ext_vector_type(8))) float v8f;

// ---------------- zero scratch (agg + BN stat accumulators) ----------------
__global__ void k_zero(float* __restrict__ p, long long n) {
    long long i = (long long)blockIdx.x * blockDim.x + threadIdx.x;
    long long stride = (long long)gridDim.x * blockDim.x;
    for (; i < n; i += stride) p[i] = 0.0f;
}

// ---------------- pack W so each lane's (W[k][n], W[k+1][n]) is contiguous --
__global__ void k_pack(const float* __restrict__ W, float* __restrict__ Wp) {
    int i = blockIdx.x * 256 + threadIdx.x;
    if (i < HH * HH) {
        int k = i >> 7, col = i & 127;
        Wp[((size_t)(k >> 1) * HH + col) * 2 + (k & 1)] = W[i];
    }
}

// ---------------- edge scatter: agg[row] += v[col], one wave per edge -------
__global__ void __launch_bounds__(256)
k_scatter(const float* __restrict__ v, const int* __restrict__ erow,
          const int* __restrict__ ecol, float* __restrict__ agg) {
    int wave = threadIdx.x >> 5;
    int lane = threadIdx.x & 31;
    int e = blockIdx.x * 8 + wave;
    if (e >= EE) return;
    int r = erow[e], c = ecol[e];
    float4 val = *(const float4*)(v + (size_t)c * HH + lane * 4);
    float* dst = agg + (size_t)r * HH + lane * 4;
    unsafeAtomicAdd(dst + 0, val.x);
    unsafeAtomicAdd(dst + 1, val.y);
    unsafeAtomicAdd(dst + 2, val.z);
    unsafeAtomicAdd(dst + 3, val.w);
}

// ---------------- fused GEMM (64 rows/block) with fp32 WMMA -----------------
// mode 0: x = X + eps*V       (GIN layer 1 input),  Y = X (in-place safe: block owns rows)
// mode 1: x = relu(a*X + c)   (BN1+ReLU fused on load), Y = d_out
// Always: Y = x @ W + bias; accumulate per-column sum / sumsq for batchnorm.
__global__ void __launch_bounds__(256)
k_gemm(const float* __restrict__ X, const float* __restrict__ V,
       const float* __restrict__ Wp, const float* __restrict__ bias,
       const float* __restrict__ abn, const float* __restrict__ cbn,
       const float* __restrict__ epsp,
       float* __restrict__ Y, float* __restrict__ colsum, float* __restrict__ colsumsq,
       int mode) {
    __shared__ float As[64][132];   // padded to dodge bank conflicts
    __shared__ float s_a[HH], s_c[HH];

    int t = threadIdx.x;
    if (mode == 1 && t < HH) { s_a[t] = abn[t]; s_c[t] = cbn[t]; }
    __syncthreads();

    float eps = (mode == 0) ? epsp[0] : 0.0f;
    long long rowBase = (long long)blockIdx.x * 64;

    // stage 64x128 input strip (with fused input transform)
    #pragma unroll
    for (int i = 0; i < 8; ++i) {
        int idx4 = t + i * 256;            // 2048 float4s
        int r = idx4 >> 5;                 // 32 float4 per row
        int c = (idx4 & 31) << 2;
        long long g = (rowBase + r) * HH + c;
        float4 x = *(const float4*)(X + g);
        if (mode == 0) {
            float4 vv = *(const float4*)(V + g);
            x.x += eps * vv.x; x.y += eps * vv.y;
            x.z += eps * vv.z; x.w += eps * vv.w;
        } else {
            x.x = fmaxf(0.0f, s_a[c + 0] * x.x + s_c[c + 0]);
            x.y = fmaxf(0.0f, s_a[c + 1] * x.y + s_c[c + 1]);
            x.z = fmaxf(0.0f, s_a[c + 2] * x.z + s_c[c + 2]);
            x.w = fmaxf(0.0f, s_a[c + 3] * x.w + s_c[c + 3]);
        }
        As[r][c + 0] = x.x; As[r][c + 1] = x.y;
        As[r][c + 2] = x.z; As[r][c + 3] = x.w;
    }
    __syncthreads();

    int w = t >> 5, lane = t & 31;
    int lo = lane & 15, hi = lane >> 4;
    int bcol = w * 16 + lo;                // wave w owns output columns [16w,16w+16)
    float bv = bias[bcol];
    float s = 0.0f, q = 0.0f;

    for (int rt = 0; rt < 4; ++rt) {
        v8f acc;
        #pragma unroll
        for (int r = 0; r < 8; ++r) acc[r] = bv;
        #pragma unroll
        for (int ks = 0; ks < 32; ++ks) {
            int kk = ks * 4 + 2 * hi;
            // A 16x4 fp32: lanes 0-15 -> K0/K1, lanes 16-31 -> K2/K3, M = lane&15
            v2f a = *(const v2f*)&As[rt * 16 + lo][kk];
            // B 4x16 fp32 mirrored: lane holds (W[kk][n], W[kk+1][n]), n = lane&15
            v2f b = *(const v2f*)(Wp + ((size_t)(kk >> 1) * HH + bcol) * 2);
            acc = __builtin_amdgcn_wmma_f32_16x16x4_f32(
                false, a, false, b, (short)0, acc, false, false);
        }
        #pragma unroll
        for (int r = 0; r < 8; ++r) {
            float val = acc[r];
            s += val; q += val * val;
            // D layout: VGPR r -> M = r + 8*hi, N = lane&15
            Y[(rowBase + rt * 16 + r + 8 * hi) * HH + bcol] = val;
        }
    }
    // lanes l and l+16 share the same output column -> fold halves
    s += __shfl_xor(s, 16, 32);
    q += __shfl_xor(q, 16, 32);
    if (lane < 16) {
        unsafeAtomicAdd(&colsum[bcol], s);
        unsafeAtomicAdd(&colsumsq[bcol], q);
    }
}

// ---------------- batchnorm finalize: a = g*rsqrt(var+eps), c = be - a*mean -
__global__ void k_finalize(const float* __restrict__ colsum, const float* __restrict__ colsumsq,
                           const float* __restrict__ g, const float* __restrict__ be,
                           float* __restrict__ a, float* __restrict__ c) {
    int i = threadIdx.x;
    if (i < HH) {
        const float invN = 1.0f / (float)NN;
        float m = colsum[i] * invN;
        float var = colsumsq[i] * invN - m * m;
        float ai = g[i] * rsqrtf(var + 1e-5f);
        a[i] = ai;
        c[i] = be[i] - ai * m;
    }
}

// ---------------- final BN2 + ReLU, in place on d_out -----------------------
__global__ void k_bnrelu(float* __restrict__ y, const float* __restrict__ a,
                         const float* __restrict__ c, long long n) {
    long long i = (long long)blockIdx.x * blockDim.x + threadIdx.x;
    long long stride = (long long)gridDim.x * blockDim.x;
    for (; i < n; i += stride) {
        int col = (int)(i & (HH - 1));
        y[i] = fmaxf(0.0f, a[col] * y[i] + c[col]);
    }
}

// ---------------- biased random walk (tiny, one block) ----------------------
__global__ void __launch_bounds__(128)
k_walk(const int* __restrict__ adj, const float* __restrict__ scores,
       const float* __restrict__ feats, const int* __restrict__ perm,
       const float* __restrict__ xsmall, float* __restrict__ xout) {
    __shared__ float s_scores[NB_];
    __shared__ unsigned char s_inperm[NB_];
    __shared__ int s_idxperm[NB_];
    __shared__ unsigned char s_visited[NB_];
    __shared__ unsigned char s_chosen[NB_];
    __shared__ int s_starts[NSTART_];
    __shared__ unsigned char s_ok[NSTART_];
    __shared__ int s_path[MAXLEN_];
    __shared__ float s_w[MAXLEN_];
    __shared__ int s_idx;
    __shared__ int s_do;

    int t = threadIdx.x;
    // copy x_small -> output region
    for (int i = t; i < PP_ * HH; i += 128) xout[i] = xsmall[i];
    if (t < NB_) {
        s_scores[t] = scores[t];
        s_inperm[t] = 0; s_idxperm[t] = 0;
        s_visited[t] = 0; s_chosen[t] = 0;
    }
    __syncthreads();

    if (t == 0) {
        // reverse loop so the FIRST matching perm index wins (argmax(perm==start))
        for (int p = PP_ - 1; p >= 0; --p) {
            int s = perm[p];
            if (s >= 0 && s < NB_) { s_inperm[s] = 1; s_idxperm[s] = p; }
        }
        // starts = top-NSTART in-perm nodes by descending score, ties -> lowest index
        for (int k = 0; k < NSTART_; ++k) {
            int best = -1;
            for (int j = 0; j < NB_; ++j)
                if (s_inperm[j] && !s_chosen[j])
                    if (best < 0 || s_scores[j] > s_scores[best]) best = j;
            if (best >= 0) { s_starts[k] = best; s_ok[k] = 1; s_chosen[best] = 1; }
            else           { s_starts[k] = 0;    s_ok[k] = 0; }
        }
    }
    __syncthreads();

    for (int k = 0; k < NSTART_; ++k) {
        if (t == 0) {
            s_do = 0;
            int start = s_starts[k];
            int skip = (!s_ok[k]) || s_visited[start];
            if (!skip) {
                s_visited[start] = 1;
                s_path[0] = start;
                int valids[MAXLEN_];
                valids[0] = 1;
                int cur = start, alive = 1;
                for (int st = 1; st < MAXLEN_; ++st) {
                    int best = -1;
                    if (alive) {
                        // argmax over softmax(scores) == argmax over scores (monotone)
                        for (int j = 0; j < NB_; ++j)
                            if (adj[cur * NB_ + j] > 0 && !s_visited[j])
                                if (best < 0 || s_scores[j] > s_scores[best]) best = j;
                    }
                    int has = (best >= 0);
                    if (has) { cur = best; s_visited[cur] = 1; }
                    s_path[st] = cur;
                    valids[st] = has;
                    alive = has;
                }
                // softmax over valid path scores
                float mx = -INFINITY;
                for (int st = 0; st < MAXLEN_; ++st)
                    if (valids[st]) { float sc = s_scores[s_path[st]]; if (sc > mx) mx = sc; }
                float e[MAXLEN_]; float sum = 0.0f;
                for (int st = 0; st < MAXLEN_; ++st) {
                    e[st] = valids[st] ? expf(s_scores[s_path[st]] - mx) : 0.0f;
                    sum += e[st];
                }
                for (int st = 0; st < MAXLEN_; ++st) s_w[st] = e[st] / sum;
                s_idx = s_idxperm[start];
                s_do = 1;
            }
        }
        __syncthreads();
        if (s_do) {
            float f = 0.0f;
            for (int st = 0; st < MAXLEN_; ++st)
                f += s_w[st] * feats[s_path[st] * HH + t];
            xout[s_idx * HH + t] += f;
        }
        __syncthreads();
    }
}

extern "C" void kernel_launch(void* const* d_in, const int* in_sizes, int n_in,
                              void* d_out, int out_size, void* d_ws, size_t ws_size,
                              hipStream_t stream) {
    const float* v     = (const float*)d_in[0];
    const int*   erow  = (const int*)d_in[1];
    const int*   ecol  = (const int*)d_in[2];
    const float* W1    = (const float*)d_in[3];
    const float* b1    = (const float*)d_in[4];
    const float* g1    = (const float*)d_in[5];
    const float* be1   = (const float*)d_in[6];
    const float* W2    = (const float*)d_in[7];
    const float* b2    = (const float*)d_in[8];
    const float* g2    = (const float*)d_in[9];
    const float* be2   = (const float*)d_in[10];
    const float* epsp  = (const float*)d_in[11];
    const int*   adj   = (const int*)d_in[12];
    const float* scrs  = (const float*)d_in[13];
    const float* feats = (const float*)d_in[14];
    const int*   perm  = (const int*)d_in[15];
    const float* xsml  = (const float*)d_in[16];
    float* out = (float*)d_out;

    float* ws       = (float*)d_ws;
    float* agg      = ws;                          // N*H, reused as y1 in-place
    float* colsum1  = ws + (size_t)NN * HH;
    float* colsq1   = colsum1 + HH;
    float* colsum2  = colsq1 + HH;
    float* colsq2   = colsum2 + HH;
    float* a1 = colsq2 + HH;  float* c1 = a1 + HH;
    float* a2 = c1 + HH;      float* c2 = a2 + HH;
    float* Wp1 = c2 + HH;
    float* Wp2 = Wp1 + HH * HH;

    // zero agg + the four 128-wide stat accumulators (contiguous)
    k_zero<<<4096, 256, 0, stream>>>(agg, (long long)NN * HH + 4 * HH);
    k_pack<<<64, 256, 0, stream>>>(W1, Wp1);
    k_pack<<<64, 256, 0, stream>>>(W2, Wp2);
    k_scatter<<<(EE + 7) / 8, 256, 0, stream>>>(v, erow, ecol, agg);

    // layer 1: y1 = (agg + eps*v) @ W1 + b1   (in place), BN1 stats
    k_gemm<<<NN / 64, 256, 0, stream>>>(agg, v, Wp1, b1, nullptr, nullptr, epsp,
                                        agg, colsum1, colsq1, 0);
    k_finalize<<<1, 128, 0, stream>>>(colsum1, colsq1, g1, be1, a1, c1);

    // layer 2: y2 = relu(a1*y1+c1) @ W2 + b2  -> d_out, BN2 stats
    k_gemm<<<NN / 64, 256, 0, stream>>>(agg, nullptr, Wp2, b2, a1, c1, epsp,
                                        out, colsum2, colsq2, 1);
    k_finalize<<<1, 128, 0, stream>>>(colsum2, colsq2, g2, be2, a2, c2);
    k_bnrelu<<<4096, 256, 0, stream>>>(out, a2, c2, (long long)NN * HH);

    // random-walk fused features -> tail of d_out
    k_walk<<<1, 128, 0, stream>>>(adj, scrs, feats, perm, xsml, out + (size_t)NN * HH);
}